// Network_1571958030343
// MI455X (gfx1250) — compile-verified
//
#include <hip/hip_runtime.h>
#include <math.h>

// ---------------------------------------------------------------------------
// CDNA5 (gfx1250) implementation of the NAR-GNN forward pass.
//
// Factorization: p[src]@Wm1 + p[dst]@Wm2  ==  A[src] + B[dst]  with
//   A = p@Wm1, B = p@Wm2 computed once per node via V_WMMA_F32_16X16X4_F32
// (f32 tensor-core path, matching reference precision). Edge phase is then
// memory-bound (fits in 192MB L2); segment_max is done with u32 atomicMax on
// float bit patterns (valid: all maxed values are >= 0, IEEE order-preserving,
// and max is order-invariant => deterministic).
// ---------------------------------------------------------------------------

#define NN 20000
#define NE 320000
#define LAT 128
#define TH 4
#define EPSF 1e-7f

typedef __attribute__((ext_vector_type(2))) float v2f;
typedef __attribute__((ext_vector_type(8))) float v8f;

// ---- Node GEMM: A = p@Wm1, B = p@Wm2 ; p[n] = [relu(inp@We+be), h[n]] -----
// One wave per (16-node tile, 16-col tile). blockDim=256 => 8 waves = 8 col
// tiles covering all 128 output columns. grid = NN/16 = 1250.
// K-loop split in two straight-line halves (no wave-uniform branching inside
// the loop), unrolled x4 => dense v_wmma_f32_16x16x4_f32 stream.
__global__ __launch_bounds__(256)
void gemm_node(const float* __restrict__ pos, const float* __restrict__ iv,
               const float* __restrict__ h,
               const float* __restrict__ We, const float* __restrict__ be,
               const float* __restrict__ Wm1, const float* __restrict__ Wm2,
               float* __restrict__ Ab, float* __restrict__ Bb)
{
    const int wave = threadIdx.x >> 5;      // col tile 0..7
    const int lane = threadIdx.x & 31;
    const int r16  = lane & 15;
    const int kh   = lane >> 4;             // 0: K pair (0,1), 1: K pair (2,3)
    const int tile = blockIdx.x;
    const int node = tile * 16 + r16;       // row for A fragment
    const int col  = wave * 16 + r16;       // column for B fragment / store

    const float pn = pos[node];
    const float yn = iv[node];
    const float* hrow = h + (size_t)node * LAT;

    v8f accA = {};
    v8f accB = {};

    // ---- Phase 1: K in [0,128) -- A elements are z = relu(inp@We+be) ------
#pragma unroll 4
    for (int kb = 0; kb < LAT; kb += 4) {
        const int k0 = kb + 2 * kh;              // lane's first K element (even)
        const float2 w0 = *(const float2*)(We + k0);         // We[0][k0..k0+1]
        const float2 w1 = *(const float2*)(We + LAT + k0);   // We[1][k0..k0+1]
        const float2 b0 = *(const float2*)(be + k0);
        v2f a;                                    // ISA A layout: v0=K(k0), v1=K(k0+1)
        a.x = fmaxf(fmaf(pn, w0.x, fmaf(yn, w1.x, b0.x)), 0.f);
        a.y = fmaxf(fmaf(pn, w0.y, fmaf(yn, w1.y, b0.y)), 0.f);

        // B 4x16 layout: VGPR0 = rows {kb, kb+2}, VGPR1 = rows {kb+1, kb+3}
        v2f b1, b2;
        b1.x = Wm1[(size_t)k0 * LAT + col];
        b1.y = Wm1[(size_t)(k0 + 1) * LAT + col];
        b2.x = Wm2[(size_t)k0 * LAT + col];
        b2.y = Wm2[(size_t)(k0 + 1) * LAT + col];

        accA = __builtin_amdgcn_wmma_f32_16x16x4_f32(false, a, false, b1,
                                                     (short)0, accA, false, false);
        accB = __builtin_amdgcn_wmma_f32_16x16x4_f32(false, a, false, b2,
                                                     (short)0, accB, false, false);
    }

    // ---- Phase 2: K in [128,256) -- A elements come from h[node] ----------
#pragma unroll 4
    for (int kb = 0; kb < LAT; kb += 4) {
        const int k0 = kb + 2 * kh;               // even => 8B-aligned b64 load
        const float2 hv = *(const float2*)(hrow + k0);
        v2f a; a.x = hv.x; a.y = hv.y;

        const int kw = LAT + k0;                  // row index into Wm1/Wm2
        v2f b1, b2;
        b1.x = Wm1[(size_t)kw * LAT + col];
        b1.y = Wm1[(size_t)(kw + 1) * LAT + col];
        b2.x = Wm2[(size_t)kw * LAT + col];
        b2.y = Wm2[(size_t)(kw + 1) * LAT + col];

        accA = __builtin_amdgcn_wmma_f32_16x16x4_f32(false, a, false, b1,
                                                     (short)0, accA, false, false);
        accB = __builtin_amdgcn_wmma_f32_16x16x4_f32(false, a, false, b2,
                                                     (short)0, accB, false, false);
    }

    // C/D layout: VGPR r -> row r (lanes 0-15) / r+8 (lanes 16-31), N = lane&15
#pragma unroll
    for (int r = 0; r < 8; ++r) {
        const int n = tile * 16 + r + kh * 8;
        Ab[(size_t)n * LAT + col] = accA[r];
        Bb[(size_t)n * LAT + col] = accB[r];
    }
}

// ---- zero helper -----------------------------------------------------------
__global__ __launch_bounds__(256)
void zero_f(float* __restrict__ p, int count)
{
    const int i = blockIdx.x * 256 + threadIdx.x;
    if (i < count) p[i] = 0.f;
}

// ---- edge message + segment_max into h[dst] --------------------------------
// 32 threads per edge, 4 columns each (coalesced 128-float rows).
__global__ __launch_bounds__(256)
void edge_msg(const int* __restrict__ src, const int* __restrict__ dst,
              const float* __restrict__ Ab, const float* __restrict__ Bb,
              const float* __restrict__ bm, unsigned int* __restrict__ hbits)
{
    const int tid = blockIdx.x * 256 + threadIdx.x;
    const int e = tid >> 5;
    if (e >= NE) return;
    const int j = (tid & 31) * 4;
    const int s = src[e], d = dst[e];
    const float4 a = *(const float4*)(Ab + (size_t)s * LAT + j);
    const float4 b = *(const float4*)(Bb + (size_t)d * LAT + j);
    const float4 c = *(const float4*)(bm + j);
    unsigned int* hp = hbits + (size_t)d * LAT + j;
    // relu => values >= 0 => u32 max on bits == float max (init 0 == 0.0f)
    atomicMax(hp + 0, __float_as_uint(fmaxf(a.x + b.x + c.x, 0.f)));
    atomicMax(hp + 1, __float_as_uint(fmaxf(a.y + b.y + c.y, 0.f)));
    atomicMax(hp + 2, __float_as_uint(fmaxf(a.z + b.z + c.z, 0.f)));
    atomicMax(hp + 3, __float_as_uint(fmaxf(a.w + b.w + c.w, 0.f)));
}

// ---- per-node decoder dots u = h@Wd[:128], v = h@Wd[128:]; zero nodemax ---
__global__ __launch_bounds__(256)
void node_uv(const float* __restrict__ h, const float* __restrict__ Wd,
             float* __restrict__ u, float* __restrict__ v,
             unsigned int* __restrict__ nodemax)
{
    const int wave = threadIdx.x >> 5, lane = threadIdx.x & 31;
    const int n = blockIdx.x * 8 + wave;
    if (n >= NN) return;
    const float* hr = h + (size_t)n * LAT;
    float su = 0.f, sv = 0.f;
#pragma unroll
    for (int q = 0; q < 4; ++q) {
        const int j = lane + 32 * q;
        const float hv = hr[j];
        su = fmaf(hv, Wd[j], su);
        sv = fmaf(hv, Wd[LAT + j], sv);
    }
#pragma unroll
    for (int off = 16; off > 0; off >>= 1) {   // wave32 tree reduce (fixed order)
        su += __shfl_down(su, off, 32);
        sv += __shfl_down(sv, off, 32);
    }
    if (lane == 0) { u[n] = su; v[n] = sv; nodemax[n] = 0u; }
}

// ---- edge decoder: alpha, node_max scatter, hint/x BCE partials ------------
__global__ __launch_bounds__(256)
void edge_alpha(const int* __restrict__ src, const int* __restrict__ dst,
                const float* __restrict__ u, const float* __restrict__ v,
                const float* __restrict__ bd,
                const float* __restrict__ hint, const float* __restrict__ edges,
                int last, float* __restrict__ alpha,
                unsigned int* __restrict__ nodemax,
                float* __restrict__ part_h, float* __restrict__ part_x)
{
    __shared__ float sh[256], sx[256];
    const int e = blockIdx.x * 256 + threadIdx.x;
    float th_ = 0.f, tx_ = 0.f;
    if (e < NE) {
        const int s = src[e], d = dst[e];
        const float x = u[s] + v[d] + bd[0];
        const float al = 1.0f / (1.0f + expf(-x));
        alpha[e] = al;
        const unsigned ab = __float_as_uint(al);   // al in (0,1) => positive
        atomicMax(nodemax + s, ab);
        atomicMax(nodemax + d, ab);
        const float pc = fminf(fmaxf(al, EPSF), 1.0f - EPSF);
        const float lp = logf(pc), lq = logf(1.0f - pc);
        const float t0 = hint[e];
        th_ = -(t0 * lp + (1.0f - t0) * lq);
        if (last) { const float t1 = edges[e]; tx_ = -(t1 * lp + (1.0f - t1) * lq); }
    }
    sh[threadIdx.x] = th_; sx[threadIdx.x] = tx_;
    __syncthreads();
    for (int s2 = 128; s2 > 0; s2 >>= 1) {
        if (threadIdx.x < s2) {
            sh[threadIdx.x] += sh[threadIdx.x + s2];
            sx[threadIdx.x] += sx[threadIdx.x + s2];
        }
        __syncthreads();
    }
    if (threadIdx.x == 0) {
        part_h[blockIdx.x] = sh[0];
        if (last) part_x[blockIdx.x] = sx[0];
    }
}

// ---- y threshold + (last iter) reach BCE partials --------------------------
__global__ __launch_bounds__(256)
void node_y(const unsigned int* __restrict__ nodemax, float* __restrict__ y,
            const float* __restrict__ reach, int last, float* __restrict__ part_r)
{
    __shared__ float sr[256];
    const int n = blockIdx.x * 256 + threadIdx.x;
    float term = 0.f;
    if (n < NN) {
        const float nm = __uint_as_float(nodemax[n]);
        const float yv = (nm >= 0.4f) ? 1.0f : 0.0f;
        y[n] = yv;
        if (last) {
            const float pc = fminf(fmaxf(yv, EPSF), 1.0f - EPSF);
            const float tr = reach[n];
            term = -(tr * logf(pc) + (1.0f - tr) * logf(1.0f - pc));
        }
    }
    sr[threadIdx.x] = term;
    __syncthreads();
    for (int s2 = 128; s2 > 0; s2 >>= 1) {
        if (threadIdx.x < s2) sr[threadIdx.x] += sr[threadIdx.x + s2];
        __syncthreads();
    }
    if (threadIdx.x == 0 && last) part_r[blockIdx.x] = sr[0];
}

// ---- parent selection ------------------------------------------------------
__global__ __launch_bounds__(256)
void parent_init(unsigned int* __restrict__ dmax, int* __restrict__ first)
{
    const int n = blockIdx.x * 256 + threadIdx.x;
    if (n < NN) { dmax[n] = 0u; first[n] = NE; }
}

__global__ __launch_bounds__(256)
void edge_dmax(const int* __restrict__ dst, const float* __restrict__ alpha,
               unsigned int* __restrict__ dmax)
{
    const int e = blockIdx.x * 256 + threadIdx.x;
    if (e < NE) atomicMax(dmax + dst[e], __float_as_uint(alpha[e]));
}

__global__ __launch_bounds__(256)
void edge_first(const int* __restrict__ dst, const float* __restrict__ alpha,
                const unsigned int* __restrict__ dmax, int* __restrict__ first)
{
    const int e = blockIdx.x * 256 + threadIdx.x;
    if (e < NE) {
        const int d = dst[e];
        if (__float_as_uint(alpha[e]) == dmax[d]) atomicMin(first + d, e);
    }
}

__global__ __launch_bounds__(256)
void node_parent(const int* __restrict__ first, const int* __restrict__ src,
                 const int* __restrict__ pi, float* __restrict__ part_p)
{
    __shared__ float sp[256];
    const int n = blockIdx.x * 256 + threadIdx.x;
    float m = 0.f;
    if (n < NN) {
        const int f = first[n];
        const int parent = (f < NE) ? src[f] : n;
        m = (parent == pi[n]) ? 1.0f : 0.0f;
    }
    sp[threadIdx.x] = m;
    __syncthreads();
    for (int s2 = 128; s2 > 0; s2 >>= 1) {
        if (threadIdx.x < s2) sp[threadIdx.x] += sp[threadIdx.x + s2];
        __syncthreads();
    }
    if (threadIdx.x == 0) part_p[blockIdx.x] = sp[0];
}

// ---- final deterministic reduction -> 4 losses -----------------------------
__global__ __launch_bounds__(256)
void finalize(const float* __restrict__ part_x, const float* __restrict__ part_h,
              const float* __restrict__ part_r, const float* __restrict__ part_p,
              float* __restrict__ out)
{
    __shared__ float s0[256], s1[256], s2[256], s3[256];
    float a = 0.f, b = 0.f, c = 0.f, d = 0.f;
    for (int i = threadIdx.x; i < 1250;     i += 256) a += part_x[i];
    for (int i = threadIdx.x; i < 4 * 1250; i += 256) b += part_h[i];
    for (int i = threadIdx.x; i < 79;       i += 256) c += part_r[i];
    for (int i = threadIdx.x; i < 79;       i += 256) d += part_p[i];
    s0[threadIdx.x] = a; s1[threadIdx.x] = b; s2[threadIdx.x] = c; s3[threadIdx.x] = d;
    __syncthreads();
    for (int s = 128; s > 0; s >>= 1) {
        if (threadIdx.x < s) {
            s0[threadIdx.x] += s0[threadIdx.x + s];
            s1[threadIdx.x] += s1[threadIdx.x + s];
            s2[threadIdx.x] += s2[threadIdx.x + s];
            s3[threadIdx.x] += s3[threadIdx.x + s];
        }
        __syncthreads();
    }
    if (threadIdx.x == 0) {
        out[0] = s0[0] / (float)NE;          // loss_x
        out[1] = s1[0] / (float)NE;          // loss_h (sum of 4 means)
        out[2] = s2[0] / (float)NN;          // loss_reach
        out[3] = 1.0f - s3[0] / (float)NN;   // loss_parents
    }
}

// ---------------------------------------------------------------------------
extern "C" void kernel_launch(void* const* d_in, const int* in_sizes, int n_in,
                              void* d_out, int out_size, void* d_ws, size_t ws_size,
                              hipStream_t stream)
{
    (void)in_sizes; (void)n_in; (void)out_size; (void)ws_size;

    const float* pos     = (const float*)d_in[0];
    const float* s       = (const float*)d_in[1];
    const float* edges   = (const float*)d_in[2];
    const float* edges_h = (const float*)d_in[3];   // (5, E)
    const float* reach_h = (const float*)d_in[4];   // (5, N)
    const float* We      = (const float*)d_in[5];   // (2, 128)
    const float* be      = (const float*)d_in[6];
    const float* Wm1     = (const float*)d_in[7];   // (256, 128)
    const float* Wm2     = (const float*)d_in[8];
    const float* bm      = (const float*)d_in[9];
    const float* Wd      = (const float*)d_in[10];  // (256, 1)
    const float* bd      = (const float*)d_in[11];
    const int*   eidx    = (const int*)d_in[12];    // (2, E)
    const int*   pi      = (const int*)d_in[13];
    const int* src = eidx;
    const int* dst = eidx + NE;

    // -------- workspace layout (all float-sized slots) --------
    float* ws = (float*)d_ws;
    size_t off = 0;
    float* hA    = ws + off; off += (size_t)NN * LAT;
    float* hB    = ws + off; off += (size_t)NN * LAT;
    float* Ab    = ws + off; off += (size_t)NN * LAT;
    float* Bb    = ws + off; off += (size_t)NN * LAT;
    float* u     = ws + off; off += NN;
    float* v     = ws + off; off += NN;
    float* alpha = ws + off; off += NE;
    unsigned int* nodemax = (unsigned int*)(ws + off); off += NN;
    float* yv    = ws + off; off += NN;
    unsigned int* dmax    = (unsigned int*)(ws + off); off += NN;
    int*   first = (int*)(ws + off); off += NN;
    float* part_x = ws + off; off += 1250;
    float* part_h = ws + off; off += 4 * 1250;
    float* part_r = ws + off; off += 128;
    float* part_p = ws + off; off += 128;

    const int NBF = (NN * LAT + 255) / 256;   // 10000
    const int NBE = NE / 256;                 // 1250
    const int NBN = (NN + 255) / 256;         // 79

    // h_0 = 0
    zero_f<<<NBF, 256, 0, stream>>>(hA, NN * LAT);

    float* hcur = hA;
    float* hnxt = hB;
    for (int t = 0; t < TH; ++t) {
        const int last = (t == TH - 1);
        const float* iv = (t == 0) ? s : yv;

        gemm_node<<<NN / 16, 256, 0, stream>>>(pos, iv, hcur, We, be, Wm1, Wm2, Ab, Bb);
        zero_f<<<NBF, 256, 0, stream>>>(hnxt, NN * LAT);
        edge_msg<<<NE * 32 / 256, 256, 0, stream>>>(src, dst, Ab, Bb, bm,
                                                    (unsigned int*)hnxt);
        node_uv<<<(NN + 7) / 8, 256, 0, stream>>>(hnxt, Wd, u, v, nodemax);
        edge_alpha<<<NBE, 256, 0, stream>>>(src, dst, u, v, bd,
                                            edges_h + (size_t)(1 + t) * NE, edges,
                                            last, alpha, nodemax,
                                            part_h + (size_t)t * 1250, part_x);
        node_y<<<NBN, 256, 0, stream>>>(nodemax, yv, reach_h + (size_t)4 * NN,
                                        last, part_r);

        float* tmp = hcur; hcur = hnxt; hnxt = tmp;
    }

    parent_init<<<NBN, 256, 0, stream>>>(dmax, first);
    edge_dmax<<<NBE, 256, 0, stream>>>(dst, alpha, dmax);
    edge_first<<<NBE, 256, 0, stream>>>(dst, alpha, dmax, first);
    node_parent<<<NBN, 256, 0, stream>>>(first, src, pi, part_p);

    finalize<<<1, 256, 0, stream>>>(part_x, part_h, part_r, part_p, (float*)d_out);
}